// PixelAugmentation_5669356834776
// MI455X (gfx1250) — compile-verified
//
#include <hip/hip_runtime.h>
#include <math.h>

// ---------------- problem geometry ----------------
#define NIMG 128          // B*T
#define CCH  3
#define HH   256
#define WW   256
#define TLW  32           // tile width  (interior)
#define TLH  16           // tile height (interior)
#define PW   (TLW + 4)    // padded width  (+/-2 halo for 5-tap blur)
#define PH   (TLH + 4)    // padded height
#define TILES_X (WW / TLW)
#define TILES_Y (HH / TLH)
#define TILES   (TILES_X * TILES_Y)
#define PADELEMS (CCH * PH * PW)   // 2160 floats per tile

#define AS1 __attribute__((address_space(1)))
#define AS3 __attribute__((address_space(3)))

// ---------------- CDNA5 async global->LDS copy ----------------
__device__ __forceinline__ void async_copy_f32(const float* g, float* l) {
#if __has_builtin(__builtin_amdgcn_global_load_async_to_lds_b32)
  __builtin_amdgcn_global_load_async_to_lds_b32(
      (AS1 int*)(__UINTPTR_TYPE__)g, (AS3 int*)l, 0, 0);
#else
  asm volatile("global_load_async_to_lds_b32 %0, %1, off"
               :: "v"((AS3 int*)l), "v"(g) : "memory");
#endif
}

__device__ __forceinline__ void wait_async0() {
#if __has_builtin(__builtin_amdgcn_s_wait_asynccnt)
  __builtin_amdgcn_s_wait_asynccnt(0);
#else
  asm volatile("s_wait_asynccnt 0x0" ::: "memory");
#endif
}

// ---------------- helpers ----------------
__device__ __forceinline__ int reflect_i(int i, int n) {
  i = (i < 0) ? -i : i;            // reflect-101, halo <= 2 so single fold ok
  i = (i >= n) ? (2 * n - 2 - i) : i;
  return i;
}

__device__ __forceinline__ float clamp01(float x) {
  return fminf(fmaxf(x, 0.0f), 1.0f);
}

// Stage reflect-padded (PH x PW x 3) tile into LDS via async loads.
__device__ __forceinline__ void load_tile_async(const float* base, int y0, int x0,
                                                float* tile) {
  for (int e = threadIdx.x; e < PADELEMS; e += 256) {
    int c  = e / (PH * PW);
    int r  = e - c * (PH * PW);
    int py = r / PW;
    int px = r - py * PW;
    int gy = reflect_i(y0 + py - 2, HH);
    int gx = reflect_i(x0 + px - 2, WW);
    async_copy_f32(base + ((size_t)c * HH + gy) * WW + gx, tile + e);
  }
  wait_async0();     // this wave's LDS writes complete
  __syncthreads();   // all waves' writes visible
}

// Selected image value (blur / sharpen / identity). py,px = interior tile
// coords (0..TLH-1, 0..TLW-1); gy,gx = global pixel coords.
__device__ __forceinline__ float compute_xsel(const float* t, int c, int py, int px,
                                              int gy, int gx, const float k[5],
                                              float sfac, float mblur, float msharp) {
  const float* tc = t + c * (PH * PW);
  float center = tc[(py + 2) * PW + (px + 2)];
  float res = center;
  if (mblur > 0.5f) {                       // separable 5x5 expanded (weights k[i]*k[j])
    float acc = 0.0f;
#pragma unroll
    for (int i = 0; i < 5; ++i) {
      const float* row = tc + (py + i) * PW + px;
      float ra = k[0]*row[0] + k[1]*row[1] + k[2]*row[2] + k[3]*row[3] + k[4]*row[4];
      acc = fmaf(k[i], ra, acc);
    }
    res = acc;
  } else if (msharp > 0.5f) {
    float deg = center;
    if (gy >= 1 && gy < HH - 1 && gx >= 1 && gx < WW - 1) {
      const float* r0 = tc + (py + 1) * PW + (px + 1);
      const float* r1 = tc + (py + 2) * PW + (px + 1);
      const float* r2 = tc + (py + 3) * PW + (px + 1);
      float s9 = r0[0] + r0[1] + r0[2] + r1[0] + r1[1] + r1[2] + r2[0] + r2[1] + r2[2];
      deg = (s9 + 4.0f * center) * (1.0f / 13.0f);
    }
    res = clamp01(sfac * center + (1.0f - sfac) * deg);
  }
  return res;
}

__device__ __forceinline__ void adjust_hue(float r, float g, float b, float hf,
                                           float* ro, float* go, float* bo) {
  float maxc = fmaxf(fmaxf(r, g), b);
  float minc = fminf(fminf(r, g), b);
  float v = maxc;
  float d = maxc - minc;
  float s = (maxc > 0.0f) ? (d / maxc) : 0.0f;
  float dd = (d > 0.0f) ? d : 1.0f;
  float rc = (maxc - r) / dd, gc = (maxc - g) / dd, bc = (maxc - b) / dd;
  float h;
  if (maxc == r)      h = bc - gc;
  else if (maxc == g) h = 2.0f + rc - bc;
  else                h = 4.0f + gc - rc;
  h = h * (1.0f / 6.0f);
  h = h - floorf(h);
  h = (d > 0.0f) ? h : 0.0f;
  h = h + hf;
  h = h - floorf(h);
  float h6 = h * 6.0f;
  float fi = floorf(h6);
  float f  = h6 - fi;
  float p  = v * (1.0f - s);
  float q  = v * (1.0f - f * s);
  float tt = v * (1.0f - (1.0f - f) * s);
  int i = ((int)fi) % 6;
  if (i < 0) i += 6;
  switch (i) {
    case 0: *ro = v;  *go = tt; *bo = p;  break;
    case 1: *ro = q;  *go = v;  *bo = p;  break;
    case 2: *ro = p;  *go = v;  *bo = tt; break;
    case 3: *ro = p;  *go = q;  *bo = v;  break;
    case 4: *ro = tt; *go = p;  *bo = v;  break;
    default:*ro = v;  *go = p;  *bo = q;  break;
  }
}

// ---------------- kernel 1: per-image params + zero mean accumulators ----------------
__global__ __launch_bounds__(NIMG) void aug_params(
    const float* __restrict__ rand_sel, const float* __restrict__ blur_u,
    const float* __restrict__ sharpen_u, const float* __restrict__ jitter_u,
    const float* __restrict__ color_u, float* __restrict__ prm,
    float* __restrict__ meansum) {
  int n = threadIdx.x;
  if (n >= NIMG) return;
  float std_ = 0.3f + blur_u[n] * (2.0f - 0.3f);
  float pdf[5], ssum = 0.0f;
#pragma unroll
  for (int i = 0; i < 5; ++i) {
    float z = ((float)i - 2.0f) / std_;
    pdf[i] = expf(-0.5f * z * z);
    ssum += pdf[i];
  }
  float* P = prm + n * 16;
#pragma unroll
  for (int i = 0; i < 5; ++i) P[i] = pdf[i] / ssum;
  P[5] = 0.5f + sharpen_u[n] * 1.5f;                 // sharpen factor
  P[6] = 1.0f - 0.4f + color_u[n * 4 + 0] * 0.8f;    // brightness
  P[7] = 1.0f - 0.4f + color_u[n * 4 + 1] * 0.8f;    // contrast
  P[8] = 1.0f - 0.4f + color_u[n * 4 + 2] * 0.8f;    // saturation
  P[9] = -0.1f + color_u[n * 4 + 3] * 0.2f;          // hue shift
  float rs = rand_sel[n];
  P[10] = (rs < 0.3f) ? 1.0f : 0.0f;                 // blur mask
  P[11] = (rs >= 0.3f && rs < 0.6f) ? 1.0f : 0.0f;   // sharpen mask
  P[12] = (jitter_u[n] < 0.8f) ? 1.0f : 0.0f;        // jitter mask
  P[13] = P[14] = P[15] = 0.0f;
  meansum[n] = 0.0f;                                  // re-zeroed every call
}

// ---------------- kernel 2: selected image -> gray-mean accumulation ----------------
__global__ __launch_bounds__(256) void aug_pass1(const float* __restrict__ img,
                                                 const float* __restrict__ prm,
                                                 float* __restrict__ meansum) {
  __shared__ float tile[PADELEMS];
  __shared__ float red[256];
  int blk = blockIdx.x;
  int n = blk / TILES;
  int t = blk - n * TILES;
  int ty = t / TILES_X, tx = t - ty * TILES_X;
  int x0 = tx * TLW, y0 = ty * TLH;
  const float* base = img + (size_t)n * CCH * HH * WW;
  load_tile_async(base, y0, x0, tile);

  const float* P = prm + n * 16;
  float k[5];
#pragma unroll
  for (int i = 0; i < 5; ++i) k[i] = P[i];
  float sfac = P[5], bf = P[6], mblur = P[10], msharp = P[11];

  int px  = threadIdx.x & (TLW - 1);
  int py0 = threadIdx.x >> 5;      // 0..7 ; each thread does rows py0, py0+8
  float acc = 0.0f;
#pragma unroll
  for (int s = 0; s < 2; ++s) {
    int py = py0 + s * 8;
    int gy = y0 + py, gx = x0 + px;
    float v0 = compute_xsel(tile, 0, py, px, gy, gx, k, sfac, mblur, msharp);
    float v1 = compute_xsel(tile, 1, py, px, gy, gx, k, sfac, mblur, msharp);
    float v2 = compute_xsel(tile, 2, py, px, gy, gx, k, sfac, mblur, msharp);
    float jr = clamp01(bf * v0), jg = clamp01(bf * v1), jb = clamp01(bf * v2);
    acc += 0.2989f * jr + 0.587f * jg + 0.114f * jb;
  }
  red[threadIdx.x] = acc;
  __syncthreads();
  for (int off = 128; off > 0; off >>= 1) {
    if (threadIdx.x < off) red[threadIdx.x] += red[threadIdx.x + off];
    __syncthreads();
  }
  if (threadIdx.x == 0) atomicAdd(&meansum[n], red[0]);
}

// ---------------- kernel 3: full pipeline + NT stores ----------------
__global__ __launch_bounds__(256) void aug_pass2(const float* __restrict__ img,
                                                 const float* __restrict__ prm,
                                                 const float* __restrict__ meansum,
                                                 float* __restrict__ out) {
  __shared__ float tile[PADELEMS];
  int blk = blockIdx.x;
  int n = blk / TILES;
  int t = blk - n * TILES;
  int ty = t / TILES_X, tx = t - ty * TILES_X;
  int x0 = tx * TLW, y0 = ty * TLH;
  const float* base = img + (size_t)n * CCH * HH * WW;
  load_tile_async(base, y0, x0, tile);

  const float* P = prm + n * 16;
  float k[5];
#pragma unroll
  for (int i = 0; i < 5; ++i) k[i] = P[i];
  float sfac = P[5], bf = P[6], cf = P[7], sf = P[8], hf = P[9];
  float mblur = P[10], msharp = P[11], mjit = P[12];
  float mean = meansum[n] * (1.0f / (float)(HH * WW));

  float* obase = out + (size_t)n * CCH * HH * WW;
  int px  = threadIdx.x & (TLW - 1);
  int py0 = threadIdx.x >> 5;
#pragma unroll
  for (int s = 0; s < 2; ++s) {
    int py = py0 + s * 8;
    int gy = y0 + py, gx = x0 + px;
    float v0 = compute_xsel(tile, 0, py, px, gy, gx, k, sfac, mblur, msharp);
    float v1 = compute_xsel(tile, 1, py, px, gy, gx, k, sfac, mblur, msharp);
    float v2 = compute_xsel(tile, 2, py, px, gy, gx, k, sfac, mblur, msharp);
    // brightness
    float jr = clamp01(bf * v0), jg = clamp01(bf * v1), jb = clamp01(bf * v2);
    // contrast (per-image gray mean)
    jr = clamp01(cf * jr + (1.0f - cf) * mean);
    jg = clamp01(cf * jg + (1.0f - cf) * mean);
    jb = clamp01(cf * jb + (1.0f - cf) * mean);
    // saturation (per-pixel gray)
    float gray = 0.2989f * jr + 0.587f * jg + 0.114f * jb;
    jr = clamp01(sf * jr + (1.0f - sf) * gray);
    jg = clamp01(sf * jg + (1.0f - sf) * gray);
    jb = clamp01(sf * jb + (1.0f - sf) * gray);
    // hue
    float hr, hg, hb;
    adjust_hue(jr, jg, jb, hf, &hr, &hg, &hb);
    float o0 = (mjit > 0.5f) ? hr : v0;
    float o1 = (mjit > 0.5f) ? hg : v1;
    float o2 = (mjit > 0.5f) ? hb : v2;
    size_t pidx = (size_t)gy * WW + gx;
    __builtin_nontemporal_store(o0, obase + pidx);
    __builtin_nontemporal_store(o1, obase + (size_t)HH * WW + pidx);
    __builtin_nontemporal_store(o2, obase + (size_t)2 * HH * WW + pidx);
  }
}

// ---------------- launch ----------------
extern "C" void kernel_launch(void* const* d_in, const int* in_sizes, int n_in,
                              void* d_out, int out_size, void* d_ws, size_t ws_size,
                              hipStream_t stream) {
  const float* image     = (const float*)d_in[0];
  const float* rand_sel  = (const float*)d_in[1];
  const float* blur_u    = (const float*)d_in[2];
  const float* sharpen_u = (const float*)d_in[3];
  const float* jitter_u  = (const float*)d_in[4];
  const float* color_u   = (const float*)d_in[5];
  float* out = (float*)d_out;

  float* prm     = (float*)d_ws;        // 128 * 16 floats
  float* meansum = prm + NIMG * 16;     // 128 floats

  aug_params<<<1, NIMG, 0, stream>>>(rand_sel, blur_u, sharpen_u, jitter_u,
                                     color_u, prm, meansum);
  int blocks = NIMG * TILES;            // 16384
  aug_pass1<<<blocks, 256, 0, stream>>>(image, prm, meansum);
  aug_pass2<<<blocks, 256, 0, stream>>>(image, prm, meansum, out);
}